// UnionRGCNLayer_14955076125444
// MI455X (gfx1250) — compile-verified
//
#include <hip/hip_runtime.h>
#include <hip/hip_bf16.h>
#include <math.h>

#define DIM     256
#define NNODES  50000
#define NEDGES  300000
#define NRELS   500
#define ROWT    (NNODES / 16)   // 3125 row tiles (exact)
#define COLT    (DIM / 16)      // 16 col tiles

typedef __attribute__((ext_vector_type(2))) float v2f;
typedef __attribute__((ext_vector_type(8))) float v8f;

__device__ __forceinline__ v8f wmma4(v2f a, v2f b, v8f c) {
    // V_WMMA_F32_16X16X4_F32 : D(16x16 f32) = A(16x4 f32) * B(4x16 f32) + C
    return __builtin_amdgcn_wmma_f32_16x16x4_f32(false, a, false, b, (short)0, c,
                                                 false, false);
}

// C[M x 256] = A[M x 256] @ W[256 x 256].
// Each wave owns an RPW x CPW super-tile of 16x16 tiles. Out-of-range row
// tiles / rows are CLAMPED for compute (loads always valid, no EXEC games)
// and guarded only at the stores.
template <int RPW, int CPW>
__global__ void __launch_bounds__(128)
gemm_wmma(const float* __restrict__ A, const float* __restrict__ W,
          float* __restrict__ C, int M, int row_tiles, int row_grps) {
    const int t = blockIdx.x * 4 + (threadIdx.x >> 5);    // wave-uniform
    const int cgrps = COLT / CPW;
    const int g  = t / cgrps;
    if (g >= row_grps) return;                            // whole wave exits
    const int cg = t % cgrps;
    const int lane = threadIdx.x & 31;
    const int half = lane >> 4;                           // K sub-pair select
    const int l    = lane & 15;

    const float* ap[RPW];
#pragma unroll
    for (int i = 0; i < RPW; ++i) {
        int rt = g * RPW + i; if (rt >= row_tiles) rt = row_tiles - 1;
        int arow = rt * 16 + l; if (arow >= M) arow = M - 1;   // clamp, no branch
        ap[i] = A + (size_t)arow * DIM + 2 * half;
    }
    const float* wp[CPW];
#pragma unroll
    for (int j = 0; j < CPW; ++j)
        wp[j] = W + (size_t)(2 * half) * DIM + (cg * CPW + j) * 16 + l;

    v8f acc[RPW][CPW] = {};
#pragma unroll 2
    for (int k = 0; k < DIM; k += 4) {
        v2f a[RPW];
#pragma unroll
        for (int i = 0; i < RPW; ++i) a[i] = *(const v2f*)(ap[i] + k);  // b64
        v2f b[CPW];
#pragma unroll
        for (int j = 0; j < CPW; ++j) {
            b[j].x = wp[j][(size_t)k * DIM];
            b[j].y = wp[j][(size_t)(k + 1) * DIM];
        }
#pragma unroll
        for (int i = 0; i < RPW; ++i)
#pragma unroll
            for (int j = 0; j < CPW; ++j)
                acc[i][j] = wmma4(a[i], b[j], acc[i][j]);  // 16 indep chains
    }

#pragma unroll
    for (int i = 0; i < RPW; ++i) {
        const int rt = g * RPW + i;
        if (rt >= row_tiles) break;                        // only owner stores
#pragma unroll
        for (int j = 0; j < CPW; ++j) {
            const int col = (cg * CPW + j) * 16 + l;
#pragma unroll
            for (int r = 0; r < 8; ++r) {
                const int crow = rt * 16 + r + 8 * half;
                if (crow < M) C[(size_t)crow * DIM + col] = acc[i][j][r];
            }
        }
    }
}

// agg[dst] += ht[src] + rel_t[etype] ; in_deg[dst]++ .  64 lanes/edge, float4.
__global__ void __launch_bounds__(256)
edge_scatter(const float* __restrict__ ht, const float* __restrict__ rel_t,
             const int* __restrict__ src, const int* __restrict__ dst,
             const int* __restrict__ ety, float* __restrict__ agg,
             int* __restrict__ in_deg) {
    const int tid = blockIdx.x * 256 + threadIdx.x;
    const int e   = tid >> 6;
    if (e >= NEDGES) return;
    const int c = (tid & 63) << 2;
    const int s = src[e], d = dst[e], r = ety[e];
    const float4 hv = *(const float4*)(ht    + (size_t)s * DIM + c);
    const float4 rv = *(const float4*)(rel_t + (size_t)r * DIM + c);
    float* ob = agg + (size_t)d * DIM + c;
    atomicAdd(ob + 0, hv.x + rv.x);        // global_atomic_add_f32 (L2-resident)
    atomicAdd(ob + 1, hv.y + rv.y);
    atomicAdd(ob + 2, hv.z + rv.z);
    atomicAdd(ob + 3, hv.w + rv.w);
    if ((tid & 63) == 0) atomicAdd(in_deg + d, 1);
}

__device__ __forceinline__ void epilogue(int rt, int col, int half,
                                         const v8f& aL, const v8f& aE,
                                         const v8f& aS, float bias,
                                         const float* __restrict__ nrm,
                                         const float* __restrict__ ph,
                                         const int* __restrict__ in_deg,
                                         float* __restrict__ out) {
#pragma unroll
    for (int r = 0; r < 8; ++r) {
        const int row = rt * 16 + r + 8 * half;
        const size_t idx = (size_t)row * DIM + col;
        const float aggv = out[idx] * nrm[row];
        const float lm   = (in_deg[row] > 0) ? aL[r] : aE[r];
        const float sw   = 1.f / (1.f + __expf(-(aS[r] + bias)));
        const float v    = sw * (aggv + lm) + (1.f - sw) * ph[idx];
        out[idx] = fmaxf(v, 0.f);
    }
}

// Fused loop/evolve/skip GEMMs + gate + ReLU. 2 row tiles per wave, shared
// h fragments across the loop and evolve matrices (10 loads : 6 wmma per k-step).
// `out` holds agg on entry; each element is read then overwritten by one lane.
__global__ void __launch_bounds__(128)
node_update(const float* __restrict__ h, const float* __restrict__ ph,
            const float* __restrict__ nrm, const float* __restrict__ Lw,
            const float* __restrict__ Ew, const float* __restrict__ Sw,
            const float* __restrict__ Sb, const int* __restrict__ in_deg,
            float* __restrict__ out) {
    const int t = blockIdx.x * 4 + (threadIdx.x >> 5);
    const int g = t >> 4;                                  // row-tile pair
    if (g >= (ROWT + 1) / 2) return;
    const int col0 = (t & 15) << 4;
    const int lane = threadIdx.x & 31;
    const int half = lane >> 4;
    const int l    = lane & 15;

    const int rt0 = g * 2;
    const int rt1 = (rt0 + 1 < ROWT) ? rt0 + 1 : ROWT - 1;  // clamp compute

    const float* ha0 = h  + (size_t)(rt0 * 16 + l) * DIM + 2 * half;
    const float* ha1 = h  + (size_t)(rt1 * 16 + l) * DIM + 2 * half;
    const float* pa0 = ph + (size_t)(rt0 * 16 + l) * DIM + 2 * half;
    const float* pa1 = ph + (size_t)(rt1 * 16 + l) * DIM + 2 * half;
    const float* lwp = Lw + (size_t)(2 * half) * DIM + col0 + l;
    const float* ewp = Ew + (size_t)(2 * half) * DIM + col0 + l;
    const float* swp = Sw + (size_t)(2 * half) * DIM + col0 + l;

    v8f aL0 = {}, aE0 = {}, aS0 = {}, aL1 = {}, aE1 = {}, aS1 = {};
#pragma unroll 2
    for (int k = 0; k < DIM; k += 4) {
        const v2f ah0 = *(const v2f*)(ha0 + k);
        const v2f ah1 = *(const v2f*)(ha1 + k);
        const v2f ap0 = *(const v2f*)(pa0 + k);
        const v2f ap1 = *(const v2f*)(pa1 + k);
        v2f bl, be, bs;
        bl.x = lwp[(size_t)k * DIM]; bl.y = lwp[(size_t)(k + 1) * DIM];
        be.x = ewp[(size_t)k * DIM]; be.y = ewp[(size_t)(k + 1) * DIM];
        bs.x = swp[(size_t)k * DIM]; bs.y = swp[(size_t)(k + 1) * DIM];
        aL0 = wmma4(ah0, bl, aL0);
        aL1 = wmma4(ah1, bl, aL1);
        aE0 = wmma4(ah0, be, aE0);
        aE1 = wmma4(ah1, be, aE1);
        aS0 = wmma4(ap0, bs, aS0);
        aS1 = wmma4(ap1, bs, aS1);
    }

    const int col   = col0 + l;
    const float bia = Sb[col];
    epilogue(rt0, col, half, aL0, aE0, aS0, bia, nrm, ph, in_deg, out);
    if (rt0 + 1 < ROWT)                                    // only owner writes
        epilogue(rt1, col, half, aL1, aE1, aS1, bia, nrm, ph, in_deg, out);
}

extern "C" void kernel_launch(void* const* d_in, const int* in_sizes, int n_in,
                              void* d_out, int out_size, void* d_ws, size_t ws_size,
                              hipStream_t stream) {
    const float* h   = (const float*)d_in[0];
    const float* ph  = (const float*)d_in[1];
    const float* er  = (const float*)d_in[2];
    const float* nrm = (const float*)d_in[3];
    const float* Wn  = (const float*)d_in[4];
    const float* Lw  = (const float*)d_in[5];
    const float* Ew  = (const float*)d_in[6];
    const float* Sw  = (const float*)d_in[7];
    const float* Sb  = (const float*)d_in[8];
    const int*   src = (const int*)d_in[9];
    const int*   dst = (const int*)d_in[10];
    const int*   ety = (const int*)d_in[11];
    float* out = (float*)d_out;

    // Workspace layout: ht[50000*256] | rel_t[512*256 padded] | in_deg[50000]
    float* htb   = (float*)d_ws;
    float* relt  = htb + (size_t)NNODES * DIM;
    int*   indeg = (int*)(relt + (size_t)512 * DIM);

    hipMemsetAsync(out,   0, (size_t)NNODES * DIM * sizeof(float), stream);
    hipMemsetAsync(indeg, 0, (size_t)NNODES * sizeof(int), stream);

    // rel_t = emb_rel @ Wn : 32 row tiles, 8 groups of 4; 4 col groups.
    {
        const int grps = 8, waves = grps * (COLT / 4);
        gemm_wmma<4, 4><<<dim3((waves + 3) / 4), dim3(128), 0, stream>>>(
            er, Wn, relt, NRELS, 32, grps);
    }
    // ht = h @ Wn : 3125 row tiles -> 782 groups of 4 (last clamped).
    {
        const int grps = (ROWT + 3) / 4, waves = grps * (COLT / 4);
        gemm_wmma<4, 4><<<dim3((waves + 3) / 4), dim3(128), 0, stream>>>(
            h, Wn, htb, NNODES, ROWT, grps);
    }
    // agg(d_out) += ht[src] + rel_t[etype]  (linearity of the edge transform)
    edge_scatter<<<dim3((NEDGES * 64) / 256), dim3(256), 0, stream>>>(
        htb, relt, src, dst, ety, out, indeg);
    // fused loop/evolve/skip GEMMs + gate + relu, in-place on d_out
    {
        const int grps = (ROWT + 1) / 2, waves = grps * COLT;
        node_update<<<dim3((waves + 3) / 4), dim3(128), 0, stream>>>(
            h, ph, nrm, Lw, Ew, Sw, Sb, indeg, out);
    }
}